// OriTripletLoss_48808008352105
// MI455X (gfx1250) — compile-verified
//
#include <hip/hip_runtime.h>
#include <hip/hip_bf16.h>

typedef float v2f __attribute__((ext_vector_type(2)));
typedef float v8f __attribute__((ext_vector_type(8)));
typedef int   v4i_t __attribute__((ext_vector_type(4)));

#define TLP_AS_G __attribute__((address_space(1)))
#define TLP_AS_L __attribute__((address_space(3)))

#define TLP_D      768          // feature dim
#define TLP_KC     32           // K chunk staged in LDS
#define TLP_NKC    (TLP_D / TLP_KC)
#define TLP_MT     128          // rows per workgroup
#define TLP_NT     128          // cols per tile iteration
#define TLP_LDSTR  36           // LDS row stride in floats (36m mod 64 distinct for m=0..15, 16B aligned)
#define TLP_MARGIN 0.3f
#define TLP_FMAX   3.402823466e38f

#if __has_builtin(__builtin_amdgcn_global_load_async_to_lds_b128) && \
    __has_builtin(__builtin_amdgcn_s_wait_asynccnt)
#define TLP_ASYNC 1
#else
#define TLP_ASYNC 0
#endif

// ---------------------------------------------------------------------------
// Kernel 1: row squared-norms + init ap/an accumulators (bit-pattern int max/min)
// ---------------------------------------------------------------------------
__global__ __launch_bounds__(256) void tlp_norms_init(const float* __restrict__ X,
                                                      float* __restrict__ sq,
                                                      int* __restrict__ apBits,
                                                      int* __restrict__ anBits,
                                                      int N) {
  const int lane = threadIdx.x & 31;
  const int wid  = threadIdx.x >> 5;
  const int row  = blockIdx.x * 8 + wid;
  if (row >= N) return;

  const float4* xr = (const float4*)(X + (size_t)row * TLP_D);
  float s = 0.0f;
  #pragma unroll
  for (int j = 0; j < TLP_D / 4 / 32; ++j) {          // 192 float4s, 6 per lane
    float4 v = xr[lane + j * 32];
    s += v.x * v.x + v.y * v.y + v.z * v.z + v.w * v.w;
  }
  #pragma unroll
  for (int off = 16; off > 0; off >>= 1)
    s += __shfl_xor(s, off, 32);

  if (lane == 0) {
    sq[row]     = s;
    apBits[row] = 0;                                   // max-pos accumulator: 0.0f
    anBits[row] = __float_as_int(TLP_FMAX);            // min-neg accumulator: "inf"
  }
}

// ---------------------------------------------------------------------------
// Kernel 2: fused X·X^T (fp32 WMMA) -> distances -> hardest pos/neg mining.
// Async-LDS double-buffered pipeline when available; register-prefetch fallback.
// ---------------------------------------------------------------------------
__global__ __launch_bounds__(256) void tlp_mine_kernel(const float* __restrict__ X,
                                                       const int*   __restrict__ tgt,
                                                       const float* __restrict__ sq,
                                                       int* __restrict__ apBits,
                                                       int* __restrict__ anBits,
                                                       int N, int tilesPerWG) {
#if TLP_ASYNC
  __shared__ float As[2][TLP_MT * TLP_LDSTR];
  __shared__ float Bs[2][TLP_NT * TLP_LDSTR];
#else
  __shared__ float As[1][TLP_MT * TLP_LDSTR];
  __shared__ float Bs[1][TLP_NT * TLP_LDSTR];
#endif

  const int tid     = threadIdx.x;
  const int lane    = tid & 31;
  const int wid     = tid >> 5;          // 0..7
  const int ln      = lane & 15;
  const int laneHi  = lane >> 4;         // 0 or 1
  const int rowBase = blockIdx.x * TLP_MT;

  // This thread's staging slots (4 x float4 per array per chunk).
  int stR[4], stC[4];
  #pragma unroll
  for (int i = 0; i < 4; ++i) {
    const int fid = tid + i * 256;                 // 0..1023
    stR[i] = fid >> 3;                             // row 0..127
    stC[i] = (fid & 7) * 4;                        // col 0..28
  }

  // Per-lane static row metadata: C/D layout => VGPR v holds M = v + 8*laneHi
  const int waveRow = rowBase + wid * 16 + laneHi * 8;
  int   rowIdx[8]; float rowSq[8]; int rowTg[8];
  #pragma unroll
  for (int v = 0; v < 8; ++v) {
    const int r = waveRow + v;
    rowIdx[v] = r; rowSq[v] = sq[r]; rowTg[v] = tgt[r];
  }

  float apv[8], anv[8];
  #pragma unroll
  for (int v = 0; v < 8; ++v) { apv[v] = 0.0f; anv[v] = TLP_FMAX; }

  const int colTile0 = blockIdx.y * tilesPerWG;
  const int aRow = wid * 16 + ln;
  const int akb  = laneHi * 2;

  for (int ct = 0; ct < tilesPerWG; ++ct) {
    const int colBase = (colTile0 + ct) * TLP_NT;
    v8f acc[8] = {};

#if TLP_ASYNC
    // ---- async double-buffered pipeline ----
    // issue chunk 0 into buffer 0
    #pragma unroll
    for (int i = 0; i < 4; ++i) {
      const float* gA = X + (size_t)(rowBase + stR[i]) * TLP_D + stC[i];
      const float* gB = X + (size_t)(colBase + stR[i]) * TLP_D + stC[i];
      __builtin_amdgcn_global_load_async_to_lds_b128(
          (TLP_AS_G v4i_t*)gA,
          (TLP_AS_L v4i_t*)&As[0][stR[i] * TLP_LDSTR + stC[i]], 0, 0);
      __builtin_amdgcn_global_load_async_to_lds_b128(
          (TLP_AS_G v4i_t*)gB,
          (TLP_AS_L v4i_t*)&Bs[0][stR[i] * TLP_LDSTR + stC[i]], 0, 0);
    }
    for (int kc = 0; kc < TLP_NKC; ++kc) {
      const int cur = kc & 1;
      __builtin_amdgcn_s_wait_asynccnt(0);   // my copies into buf[cur] landed
      __syncthreads();                        // everyone's reads of buf[cur^1] drained
      if (kc + 1 < TLP_NKC) {
        const int k1 = (kc + 1) * TLP_KC;
        #pragma unroll
        for (int i = 0; i < 4; ++i) {
          const float* gA = X + (size_t)(rowBase + stR[i]) * TLP_D + k1 + stC[i];
          const float* gB = X + (size_t)(colBase + stR[i]) * TLP_D + k1 + stC[i];
          __builtin_amdgcn_global_load_async_to_lds_b128(
              (TLP_AS_G v4i_t*)gA,
              (TLP_AS_L v4i_t*)&As[cur ^ 1][stR[i] * TLP_LDSTR + stC[i]], 0, 0);
          __builtin_amdgcn_global_load_async_to_lds_b128(
              (TLP_AS_G v4i_t*)gB,
              (TLP_AS_L v4i_t*)&Bs[cur ^ 1][stR[i] * TLP_LDSTR + stC[i]], 0, 0);
        }
      }
      // compute on buf[cur] while async engine fills buf[cur^1]
      #pragma unroll
      for (int kk = 0; kk < TLP_KC; kk += 4) {
        v2f a;
        a.x = As[cur][aRow * TLP_LDSTR + kk + akb];
        a.y = As[cur][aRow * TLP_LDSTR + kk + akb + 1];
        #pragma unroll
        for (int t = 0; t < 8; ++t) {
          const int bRow = t * 16 + ln;
          v2f b;
          b.x = Bs[cur][bRow * TLP_LDSTR + kk + laneHi];
          b.y = Bs[cur][bRow * TLP_LDSTR + kk + laneHi + 2];
          acc[t] = __builtin_amdgcn_wmma_f32_16x16x4_f32(
              false, a, false, b, (short)0, acc[t], false, false);
        }
      }
    }
#else
    // ---- register-prefetch fallback pipeline ----
    float4 pa[4], pb[4];
    #pragma unroll
    for (int i = 0; i < 4; ++i) {
      pa[i] = *(const float4*)(X + (size_t)(rowBase + stR[i]) * TLP_D + stC[i]);
      pb[i] = *(const float4*)(X + (size_t)(colBase + stR[i]) * TLP_D + stC[i]);
    }
    for (int kc = 0; kc < TLP_NKC; ++kc) {
      __syncthreads();                        // previous compute done before overwrite
      #pragma unroll
      for (int i = 0; i < 4; ++i) {
        *(float4*)(&As[0][stR[i] * TLP_LDSTR + stC[i]]) = pa[i];
        *(float4*)(&Bs[0][stR[i] * TLP_LDSTR + stC[i]]) = pb[i];
      }
      __syncthreads();
      if (kc + 1 < TLP_NKC) {
        const int k1 = (kc + 1) * TLP_KC;
        #pragma unroll
        for (int i = 0; i < 4; ++i) {        // global loads overlap the WMMAs below
          pa[i] = *(const float4*)(X + (size_t)(rowBase + stR[i]) * TLP_D + k1 + stC[i]);
          pb[i] = *(const float4*)(X + (size_t)(colBase + stR[i]) * TLP_D + k1 + stC[i]);
        }
      }
      #pragma unroll
      for (int kk = 0; kk < TLP_KC; kk += 4) {
        v2f a;
        a.x = As[0][aRow * TLP_LDSTR + kk + akb];
        a.y = As[0][aRow * TLP_LDSTR + kk + akb + 1];
        #pragma unroll
        for (int t = 0; t < 8; ++t) {
          const int bRow = t * 16 + ln;
          v2f b;
          b.x = Bs[0][bRow * TLP_LDSTR + kk + laneHi];
          b.y = Bs[0][bRow * TLP_LDSTR + kk + laneHi + 2];
          acc[t] = __builtin_amdgcn_wmma_f32_16x16x4_f32(
              false, a, false, b, (short)0, acc[t], false, false);
        }
      }
    }
#endif

    // Epilogue: Gram tile -> distances -> masked max/min mining.
    #pragma unroll
    for (int t = 0; t < 8; ++t) {
      const int   col = colBase + t * 16 + ln;
      const float cs  = sq[col];
      const int   ctg = tgt[col];
      #pragma unroll
      for (int v = 0; v < 8; ++v) {
        const float g  = acc[t][v];
        const float d2 = rowSq[v] + cs - 2.0f * g;
        const float d  = sqrtf(fmaxf(d2, 1e-12f));
        if (rowTg[v] == ctg) {
          if (col != rowIdx[v]) apv[v] = fmaxf(apv[v], d);   // hardest positive
        } else {
          anv[v] = fminf(anv[v], d);                         // hardest negative
        }
      }
    }
  }

  // Column reduction across each 16-lane half, then cross-WG monotone int atomics.
  #pragma unroll
  for (int v = 0; v < 8; ++v) {
    #pragma unroll
    for (int off = 1; off < 16; off <<= 1) {
      apv[v] = fmaxf(apv[v], __shfl_xor(apv[v], off, 32));
      anv[v] = fminf(anv[v], __shfl_xor(anv[v], off, 32));
    }
    if (ln == 0) {
      atomicMax(&apBits[rowIdx[v]], __float_as_int(apv[v]));
      atomicMin(&anBits[rowIdx[v]], __float_as_int(anv[v]));
    }
  }
}

// ---------------------------------------------------------------------------
// Kernel 3: loss = mean(relu(ap - an + margin)); NaN/Inf -> 0. Single block.
// ---------------------------------------------------------------------------
__global__ __launch_bounds__(256) void tlp_finalize(const int* __restrict__ apBits,
                                                    const int* __restrict__ anBits,
                                                    float* __restrict__ out, int N) {
  __shared__ float red[8];
  const int tid  = threadIdx.x;
  const int lane = tid & 31;
  const int wid  = tid >> 5;

  float s = 0.0f;
  for (int i = tid; i < N; i += 256) {
    const float ap = __int_as_float(apBits[i]);
    const float an = __int_as_float(anBits[i]);   // FLT_MAX sentinel => relu(...)=0 (the +inf case)
    const float l  = ap - an + TLP_MARGIN;
    s += (l > 0.0f) ? l : 0.0f;
  }
  #pragma unroll
  for (int off = 16; off > 0; off >>= 1)
    s += __shfl_xor(s, off, 32);
  if (lane == 0) red[wid] = s;
  __syncthreads();
  if (wid == 0) {
    float t = (lane < 8) ? red[lane] : 0.0f;
    #pragma unroll
    for (int off = 4; off > 0; off >>= 1)
      t += __shfl_xor(t, off, 32);
    if (lane == 0) {
      float loss = t / (float)N;
      if (isnan(loss) || isinf(loss)) loss = 0.0f;
      out[0] = loss;
    }
  }
}

// ---------------------------------------------------------------------------
extern "C" void kernel_launch(void* const* d_in, const int* in_sizes, int n_in,
                              void* d_out, int out_size, void* d_ws, size_t ws_size,
                              hipStream_t stream) {
  const float* X   = (const float*)d_in[0];
  const int*   tgt = (const int*)d_in[1];
  float*       out = (float*)d_out;

  const int N = in_sizes[1];            // 8192
  float* sq     = (float*)d_ws;
  int*   apBits = (int*)(sq + N);
  int*   anBits = apBits + N;

  tlp_norms_init<<<N / 8, 256, 0, stream>>>(X, sq, apBits, anBits, N);

  const int nColTiles = N / TLP_NT;     // 64
  const int colSplits = 8;              // 512 workgroups
  const int tilesPerWG = nColTiles / colSplits;
  dim3 grid(N / TLP_MT, colSplits);
  tlp_mine_kernel<<<grid, 256, 0, stream>>>(X, tgt, sq, apBits, anBits, N, tilesPerWG);

  tlp_finalize<<<1, 256, 0, stream>>>(apBits, anBits, out, N);
}